// WaveletEnhancement_32126355374914
// MI455X (gfx1250) — compile-verified
//
#include <hip/hip_runtime.h>

// ---------------------------------------------------------------------------
// db4 2-level 2D DWT -> soft-threshold enhance -> IDWT, fp32, gfx1250.
// Each separable pass is a banded matmul on the matrix pipe:
// 6-step V_WMMA_F32_16X16X4_F32 chains fed from LDS (pitch-17, conflict-free
// ds_load_2addr pairs). LDS is filled with GLOBAL_LOAD_ASYNC_TO_LDS_B32
// (SADDR form: uniform SGPR base + 32-bit voffset, ASYNCcnt-tracked) with the
// pywt 'symmetric' extension folded into the address math. All global
// accesses use 32-bit indices off uniform bases (buffers < 2 GB) so stores
// select the SADDR form too.
// ---------------------------------------------------------------------------

typedef float v2f __attribute__((ext_vector_type(2)));
typedef float v8f __attribute__((ext_vector_type(8)));

// rec_lo (pywt db4) and derived filters:
// dec_lo[u] = rec_lo[7-u];  dec_hi[u] = (-1)^(u+1) rec_lo[u];
// rec_hi[i] = (-1)^i rec_lo[7-i]
__constant__ float c_dec_lo[8] = {
    -0.010597401784997278f,  0.032883011666982945f,  0.030841381835986965f,
    -0.18703481171888114f,  -0.02798376941698385f,   0.6308807679295904f,
     0.7148465705525415f,    0.23037781330885523f };
__constant__ float c_dec_hi[8] = {
    -0.23037781330885523f,   0.7148465705525415f,   -0.6308807679295904f,
    -0.02798376941698385f,   0.18703481171888114f,   0.030841381835986965f,
    -0.032883011666982945f, -0.010597401784997278f };
__constant__ float c_rec_lo[8] = {
     0.23037781330885523f,   0.7148465705525415f,    0.6308807679295904f,
    -0.02798376941698385f,  -0.18703481171888114f,   0.030841381835986965f,
     0.032883011666982945f, -0.010597401784997278f };
__constant__ float c_rec_hi[8] = {
    -0.010597401784997278f, -0.032883011666982945f,  0.030841381835986965f,
     0.18703481171888114f,  -0.02798376941698385f,  -0.6308807679295904f,
     0.7148465705525415f,   -0.23037781330885523f };

// gfx1250 async global->LDS copy (ASYNCcnt-tracked), SADDR form:
// copies 4 bytes from global address sbase + voffByte to the LDS byte address in ldsByte
__device__ __forceinline__ void async_ld_b32(unsigned ldsByte, int voffByte,
                                             const float* sbase) {
    asm volatile("global_load_async_to_lds_b32 %0, %1, %2"
                 :: "v"(ldsByte), "v"(voffByte), "s"(sbase)
                 : "memory");
}
__device__ __forceinline__ void wait_async_all() {
    asm volatile("s_wait_asynccnt 0x0" ::: "memory");
}
__device__ __forceinline__ unsigned abs_bits(float v) {
    return __float_as_uint(v) & 0x7fffffffu;   // |v| as monotonic u32
}
__device__ __forceinline__ unsigned umax2(unsigned a, unsigned b) {
    return a > b ? a : b;
}

__global__ __launch_bounds__(256) void k_zero(float* p, int n) {
    int i = blockIdx.x * blockDim.x + threadIdx.x;
    if (i < n) p[i] = 0.0f;
}

// Analysis band matrix: row r = 2j+b (j = output pos in tile, b = 0 lo / 1 hi)
// A[r][k] = dec_b[2j + 7 - k], nonzero for k in [2j, 2j+7].
__device__ __forceinline__ float afwd(int r, int k) {
    int u = 2 * (r >> 1) + 7 - k;
    float v = 0.0f;
    if (u >= 0 && u <= 7) v = (r & 1) ? c_dec_hi[u] : c_dec_lo[u];
    return v;
}

// Synthesis band matrix (interleaved lo/hi input rows k = 2j + ch):
// r even: A[r][r+2d+ch]   = rec_ch[6-2d], d=0..3
// r odd : A[r][r-1+2d+ch] = rec_ch[7-2d]
__device__ __forceinline__ float ainv(int r, int kk) {
    int u = (r & 1) ? (kk - r + 1) : (kk - r);
    float v = 0.0f;
    if (u >= 0 && u <= 7) {
        int idx = ((r & 1) ? 7 : 6) - ((u >> 1) << 1);
        v = (u & 1) ? c_rec_hi[idx] : c_rec_lo[idx];
    }
    return v;
}

// ---------------------------------------------------------------------------
// Forward pass: input (P, S) contiguous along S, DWT along S with symmetric
// extension; outputs lo/hi each (Sc, P) transposed, Sc = (S+7)/2.
// Optionally tracks per-image max|coeff| per band (bit-pattern atomicMax).
// ---------------------------------------------------------------------------
__global__ __launch_bounds__(128) void k_dwt_fwd(
    const float* __restrict__ in, float* __restrict__ lo, float* __restrict__ hi,
    int P, int S, int inImgStride, int outImgStride,
    float* __restrict__ loMax, float* __restrict__ hiMax)
{
    extern __shared__ float lds[];
    const int img = blockIdx.y;
    const int p0  = blockIdx.x << 4;
    const int Sc  = (S + 7) >> 1;
    const int ntiles = (Sc + 7) >> 3;        // 8 output positions per tile
    const int Emax   = 16 * ntiles + 8;      // staged extended rows (padded)
    const int E      = S + 13;               // symmetric-extended length
    const float* __restrict__ inp = in + (unsigned)(img * inImgStride);
    const int tid = threadIdx.x;

    // Async-stage the extended strip: lds[e*17 + c] = xe[e] of row p0+c.
    // Base pointer is wave-uniform (c = tid>>5) -> SGPR base, 32-bit voffset.
    {
        const int eL = tid & 31;
        for (int c = tid >> 5; c < 16; c += 4) {
            int pr = p0 + c; if (pr > P - 1) pr = P - 1;
            const float* __restrict__ rowp = inp + (unsigned)(pr * S);
            for (int e = eL; e < Emax; e += 32) {
                if (e < E) {
                    int g = (e < 6) ? (5 - e) : ((e <= S + 5) ? (e - 6) : (2 * S + 5 - e));
                    async_ld_b32(4u * (unsigned)(e * 17 + c), 4 * g, rowp);
                } else {
                    lds[e * 17 + c] = 0.0f;   // padded chain tail
                }
            }
        }
        wait_async_all();
    }
    __syncthreads();

    const int lane = tid & 31;
    const int wid  = tid >> 5;
    const int half = lane >> 4;              // K-pair selector in A/B layouts
    const int nrow = lane & 15;              // M for A, N for B/D
    const bool perpOK = (p0 + nrow) < P;

    // Tile-invariant analysis matrix, 6 chunks of 16x4.
    v2f amat[6];
#pragma unroll
    for (int s = 0; s < 6; ++s) {
        int k = 4 * s + 2 * half;
        amat[s].x = afwd(nrow, k);
        amat[s].y = afwd(nrow, k + 1);
    }

    const unsigned outBase = (unsigned)(img * outImgStride + p0 + nrow);
    unsigned mxLo = 0u, mxHi = 0u;

    for (int t = wid; t < ntiles; t += 4) {
        const int qb = t << 4;               // q0 = 2*i0
        v8f acc = {0.0f,0.0f,0.0f,0.0f,0.0f,0.0f,0.0f,0.0f};
#pragma unroll
        for (int s = 0; s < 6; ++s) {
            int row = qb + 4 * s + 2 * half;
            v2f b;
            b.x = lds[row * 17 + nrow];
            b.y = lds[(row + 1) * 17 + nrow];
            acc = __builtin_amdgcn_wmma_f32_16x16x4_f32(
                false, amat[s], false, b, (short)0, acc, false, false);
        }
        // D row M = r + 8*half -> output index i = i0 + 4*half + j, j = r>>1,
        // band = r&1. Uniform full/tail split; 32-bit SADDR-style indexing.
        const int ibase = (t << 3) + 4 * half;
        unsigned idx = outBase + (unsigned)(ibase * P);
        if (((t << 3) + 8) <= Sc) {          // wave-uniform: full tile
            if (perpOK) {
#pragma unroll
                for (int j = 0; j < 4; ++j) {
                    float vl = acc[2 * j], vh = acc[2 * j + 1];
                    lo[idx] = vl; hi[idx] = vh;
                    mxLo = umax2(mxLo, abs_bits(vl));
                    mxHi = umax2(mxHi, abs_bits(vh));
                    idx += (unsigned)P;
                }
            }
        } else if (perpOK) {                 // tail tile
#pragma unroll
            for (int j = 0; j < 4; ++j) {
                if ((ibase + j) < Sc) {
                    float vl = acc[2 * j], vh = acc[2 * j + 1];
                    lo[idx] = vl; hi[idx] = vh;
                    mxLo = umax2(mxLo, abs_bits(vl));
                    mxHi = umax2(mxHi, abs_bits(vh));
                }
                idx += (unsigned)P;
            }
        }
    }

    if (loMax != nullptr || hiMax != nullptr) {
#pragma unroll
        for (int off = 16; off > 0; off >>= 1) {
            mxLo = umax2(mxLo, (unsigned)__shfl_xor((int)mxLo, off, 32));
            mxHi = umax2(mxHi, (unsigned)__shfl_xor((int)mxHi, off, 32));
        }
        if (lane == 0) {
            if (loMax) atomicMax((unsigned int*)(loMax + img), mxLo);
            if (hiMax) atomicMax((unsigned int*)(hiMax + img), mxHi);
        }
    }
}

// ---------------------------------------------------------------------------
// Inverse pass: lo/hi each (P, n) contiguous along n -> out (2n-6, P)
// (transposed). Optional enhancement (x1.5 then soft threshold with
// t = 0.3*max|d|) applied in-place in LDS after the async fill.
// ---------------------------------------------------------------------------
__global__ __launch_bounds__(128) void k_idwt(
    const float* __restrict__ lo, const float* __restrict__ hi, float* __restrict__ out,
    int P, int n, int bandImgStride, int outImgStride,
    const float* __restrict__ loMax, const float* __restrict__ hiMax)
{
    extern __shared__ float lds[];
    const int img = blockIdx.y;
    const int p0  = blockIdx.x << 4;
    const int outLen = 2 * n - 6;
    const int ntiles = (outLen + 15) >> 4;
    const int Kpad   = 16 * ntiles + 8;
    const bool enhLo = (loMax != nullptr), enhHi = (hiMax != nullptr);
    const float thrLo = enhLo ? 0.3f * loMax[img] : 0.0f;
    const float thrHi = enhHi ? 0.3f * hiMax[img] : 0.0f;
    const float* __restrict__ loI = lo + (unsigned)(img * bandImgStride);
    const float* __restrict__ hiI = hi + (unsigned)(img * bandImgStride);
    const int tid = threadIdx.x;
    const int jL  = tid & 31;

    // Async-stage interleaved coefficients, one band at a time so the global
    // base stays wave-uniform (SGPR) and lanes sweep j contiguously:
    // LDS row k = 2j+ch holds band ch, coefficient j, column c.
    for (int c = tid >> 5; c < 16; c += 4) {
        int pr = p0 + c; if (pr > P - 1) pr = P - 1;
        const float* __restrict__ lrow = loI + (unsigned)(pr * n);
        const float* __restrict__ hrow = hiI + (unsigned)(pr * n);
        for (int j = jL; j < n; j += 32) {
            async_ld_b32(4u * (unsigned)((2 * j) * 17 + c),     4 * j, lrow);
            async_ld_b32(4u * (unsigned)((2 * j + 1) * 17 + c), 4 * j, hrow);
        }
        for (int k = 2 * n + jL; k < Kpad; k += 32)
            lds[k * 17 + c] = 0.0f;           // padded chain tail
    }
    wait_async_all();
    __syncthreads();

    // Enhancement applied in-place in LDS (all branches wave-uniform).
    if (enhLo || enhHi) {
        for (int c = tid >> 5; c < 16; c += 4) {
            if (enhLo) {
                for (int j = jL; j < n; j += 32) {
                    float e = 1.5f * lds[(2 * j) * 17 + c];
                    lds[(2 * j) * 17 + c] =
                        (e > thrLo) ? (e - thrLo) : ((e < -thrLo) ? (e + thrLo) : 0.0f);
                }
            }
            if (enhHi) {
                for (int j = jL; j < n; j += 32) {
                    float e = 1.5f * lds[(2 * j + 1) * 17 + c];
                    lds[(2 * j + 1) * 17 + c] =
                        (e > thrHi) ? (e - thrHi) : ((e < -thrHi) ? (e + thrHi) : 0.0f);
                }
            }
        }
        __syncthreads();
    }

    const int lane = tid & 31;
    const int wid  = tid >> 5;
    const int half = lane >> 4;
    const int nrow = lane & 15;
    const bool perpOK = (p0 + nrow) < P;

    v2f amat[6];
#pragma unroll
    for (int s = 0; s < 6; ++s) {
        int kk = 4 * s + 2 * half;
        amat[s].x = ainv(nrow, kk);
        amat[s].y = ainv(nrow, kk + 1);
    }

    const unsigned outBase = (unsigned)(img * outImgStride + p0 + nrow);

    for (int t = wid; t < ntiles; t += 4) {
        const int r0 = t << 4;
        v8f acc = {0.0f,0.0f,0.0f,0.0f,0.0f,0.0f,0.0f,0.0f};
#pragma unroll
        for (int s = 0; s < 6; ++s) {
            int row = r0 + 4 * s + 2 * half;
            v2f b;
            b.x = lds[row * 17 + nrow];
            b.y = lds[(row + 1) * 17 + nrow];
            acc = __builtin_amdgcn_wmma_f32_16x16x4_f32(
                false, amat[s], false, b, (short)0, acc, false, false);
        }
        const int rbase = r0 + 8 * half;
        unsigned idx = outBase + (unsigned)(rbase * P);
        if ((r0 + 16) <= outLen) {            // wave-uniform: full tile
            if (perpOK) {
#pragma unroll
                for (int r = 0; r < 8; ++r) { out[idx] = acc[r]; idx += (unsigned)P; }
            }
        } else if (perpOK) {                  // tail tile
#pragma unroll
            for (int r = 0; r < 8; ++r) {
                if ((rbase + r) < outLen) out[idx] = acc[r];
                idx += (unsigned)P;
            }
        }
    }
}

// ---------------------------------------------------------------------------
extern "C" void kernel_launch(void* const* d_in, const int* in_sizes, int n_in,
                              void* d_out, int out_size, void* d_ws, size_t ws_size,
                              hipStream_t stream) {
    (void)in_sizes; (void)n_in; (void)out_size; (void)ws_size;
    const float* x = (const float*)d_in[0];
    float* out = (float*)d_out;
    float* ws  = (float*)d_ws;

    const int NI = 32 * 3;                 // 96 independent images
    const int H = 510, W = 510, C1 = 258, C2 = 132;
    const int szL1 = C1 * H;               // level-1 single-axis strip
    const int szB1 = C1 * C1;              // level-1 subband
    const int szL2 = C2 * C1;              // level-2 single-axis strip
    const int szB2 = C2 * C2;              // level-2 subband

    // Workspace layout (floats). Region A [A0 .. A0+2*NI*szL1) is reused.
    float* mx  = ws;                       // 6 bands * NI maxima ([band][img])
    const long long A0 = 1024;
    float* lo1 = ws + A0;
    float* hi1 = lo1 + (long long)NI * szL1;
    float* aa  = hi1 + (long long)NI * szL1;   // region B: persistent L1 bands
    float* da  = aa  + (long long)NI * szB1;
    float* ad  = da  + (long long)NI * szB1;
    float* dd  = ad  + (long long)NI * szB1;
    // level-2 temporaries overlay the (dead) lo1/hi1 region
    float* lo2 = ws + A0;
    float* hi2 = lo2 + (long long)NI * szL2;
    float* aa2 = hi2 + (long long)NI * szL2;
    float* da2 = aa2 + (long long)NI * szB2;
    float* ad2 = da2 + (long long)NI * szB2;
    float* dd2 = ad2 + (long long)NI * szB2;
    float* s2a = dd2 + (long long)NI * szB2;
    float* s2b = s2a + (long long)NI * szL2;
    float* a1r = aa;                       // aa is dead after lo2/hi2 pass
    float* t1  = ws + A0;                  // level-2 temps dead by now
    float* t2  = t1 + (long long)NI * szL1;

    auto fwdLds  = [](int S)->size_t { int Sc=(S+7)>>1; int nt=(Sc+7)>>3; return (size_t)(16*nt+8)*17*sizeof(float); };
    auto invLds  = [](int n)->size_t { int oL=2*n-6; int nt=(oL+15)>>4; return (size_t)(16*nt+8)*17*sizeof(float); };
    auto blocksP = [](int P)->unsigned  { return (unsigned)((P + 15) / 16); };

    dim3 blk(128, 1, 1);

    k_zero<<<dim3(3,1,1), dim3(256,1,1), 0, stream>>>(mx, 6 * NI);

    // ---- analysis ----
    // L1 along W: x (H,W) -> lo1/hi1 (C1,H)
    k_dwt_fwd<<<dim3(blocksP(H), NI), blk, fwdLds(W), stream>>>(
        x, lo1, hi1, H, W, H * W, szL1, nullptr, nullptr);
    // L1 along H: lo1 -> aa/da (da = h1, enhanced)
    k_dwt_fwd<<<dim3(blocksP(C1), NI), blk, fwdLds(H), stream>>>(
        lo1, aa, da, C1, H, szL1, szB1, nullptr, mx + 0 * NI);
    // hi1 -> ad/dd (ad = v1, dd = d1, both enhanced)
    k_dwt_fwd<<<dim3(blocksP(C1), NI), blk, fwdLds(H), stream>>>(
        hi1, ad, dd, C1, H, szL1, szB1, mx + 1 * NI, mx + 2 * NI);
    // L2 along W: aa -> lo2/hi2 (C2,C1)
    k_dwt_fwd<<<dim3(blocksP(C1), NI), blk, fwdLds(C1), stream>>>(
        aa, lo2, hi2, C1, C1, szB1, szL2, nullptr, nullptr);
    // L2 along H: lo2 -> aa2/da2 (da2 = h2)
    k_dwt_fwd<<<dim3(blocksP(C2), NI), blk, fwdLds(C1), stream>>>(
        lo2, aa2, da2, C2, C1, szL2, szB2, nullptr, mx + 3 * NI);
    // hi2 -> ad2/dd2 (v2, d2)
    k_dwt_fwd<<<dim3(blocksP(C2), NI), blk, fwdLds(C1), stream>>>(
        hi2, ad2, dd2, C2, C1, szL2, szB2, mx + 4 * NI, mx + 5 * NI);

    // ---- synthesis (W-axis first, then H-axis; separable => order commutes) ----
    // L2: SynW(aa2, ad2s) and SynW(da2s, dd2s), then SynH -> a1r
    k_idwt<<<dim3(blocksP(C2), NI), blk, invLds(C2), stream>>>(
        aa2, ad2, s2a, C2, C2, szB2, szL2, nullptr, mx + 4 * NI);
    k_idwt<<<dim3(blocksP(C2), NI), blk, invLds(C2), stream>>>(
        da2, dd2, s2b, C2, C2, szB2, szL2, mx + 3 * NI, mx + 5 * NI);
    k_idwt<<<dim3(blocksP(C1), NI), blk, invLds(C2), stream>>>(
        s2a, s2b, a1r, C1, C2, szL2, szB1, nullptr, nullptr);
    // L1: SynW(a1r, v1s) and SynW(h1s, d1s), then SynH -> out
    k_idwt<<<dim3(blocksP(C1), NI), blk, invLds(C1), stream>>>(
        a1r, ad, t1, C1, C1, szB1, szL1, nullptr, mx + 1 * NI);
    k_idwt<<<dim3(blocksP(C1), NI), blk, invLds(C1), stream>>>(
        da, dd, t2, C1, C1, szB1, szL1, mx + 0 * NI, mx + 2 * NI);
    k_idwt<<<dim3(blocksP(W), NI), blk, invLds(C1), stream>>>(
        t1, t2, out, W, C1, szL1, H * W, nullptr, nullptr);
}